// OrthogonalLinear_24086176596504
// MI455X (gfx1250) — compile-verified
//
#include <hip/hip_runtime.h>
#include <hip/hip_bf16.h>

typedef float v2f __attribute__((ext_vector_type(2)));
typedef float v8f __attribute__((ext_vector_type(8)));

// ---------------------------------------------------------------------------
// Kernel 1: normalize reflection vectors. Block i handles vector i.
// VN is padded to 32 rows; rows >= nrefl are zero.
// ---------------------------------------------------------------------------
__global__ void k_normalize(const float* __restrict__ vec, float* __restrict__ VN,
                            int dim, int nrefl) {
    __shared__ float red[256];
    const int i = blockIdx.x;           // 0..31
    float ss = 0.0f;
    if (i < nrefl) {
        for (int k = threadIdx.x; k < dim; k += 256) {
            float v = vec[(size_t)i * dim + k];
            ss += v * v;
        }
    }
    red[threadIdx.x] = ss;
    __syncthreads();
    for (int s = 128; s > 0; s >>= 1) {
        if (threadIdx.x < s) red[threadIdx.x] += red[threadIdx.x + s];
        __syncthreads();
    }
    const float inv = 1.0f / (sqrtf(red[0]) + 1e-8f);
    for (int k = threadIdx.x; k < dim; k += 256) {
        VN[(size_t)i * dim + k] = (i < nrefl) ? vec[(size_t)i * dim + k] * inv : 0.0f;
    }
}

// ---------------------------------------------------------------------------
// Kernel 2: Gram matrix G[i][j] = vn_i . vn_j  (one block per (i,j) pair)
// ---------------------------------------------------------------------------
__global__ void k_gram(const float* __restrict__ VN, float* __restrict__ G,
                       int dim, int nrefl) {
    __shared__ float red[256];
    const int i = blockIdx.x / nrefl;
    const int j = blockIdx.x % nrefl;
    float s = 0.0f;
    for (int k = threadIdx.x; k < dim; k += 256)
        s += VN[(size_t)i * dim + k] * VN[(size_t)j * dim + k];
    red[threadIdx.x] = s;
    __syncthreads();
    for (int t = 128; t > 0; t >>= 1) {
        if (threadIdx.x < t) red[threadIdx.x] += red[threadIdx.x + t];
        __syncthreads();
    }
    if (threadIdx.x == 0) G[i * 32 + j] = red[0];
}

// ---------------------------------------------------------------------------
// Kernel 3: build compact-WY T (32x32, zero padded).
// T[i][i] = 2 ; T[r][i] = -2 * sum_{m=r}^{i-1} T[r][m] * G[m][i]  (r < i)
// Thread r owns row r; only reads its own row + global G.
// ---------------------------------------------------------------------------
__global__ void k_buildT(const float* __restrict__ G, float* __restrict__ T,
                         int nrefl) {
    const int r = threadIdx.x;          // 0..31
    for (int j = 0; j < 32; ++j) T[r * 32 + j] = 0.0f;
    __syncthreads();
    for (int i = 0; i < nrefl; ++i) {
        if (r == i) T[r * 32 + i] = 2.0f;
        __syncthreads();
        if (r < i) {
            float acc = 0.0f;
            for (int m = r; m < i; ++m) acc += T[r * 32 + m] * G[m * 32 + i];
            T[r * 32 + i] = -2.0f * acc;
        }
        __syncthreads();
    }
}

// ---------------------------------------------------------------------------
// Kernel 4: pre-swizzle VN into WMMA B-fragment layouts (float2 per lane).
// ISA f32 A/B frag: lane L -> {row/col = L&15, kbase = 2*(L>>4)}, v0=K kbase,
// v1=K kbase+1.
//   BP0[t][L] = {VN[L&15 ][4t+2h], VN[L&15 ][4t+2h+1]}   (h = L>>4), t<dim/4
//   BP1[t][L] = same with columns 16..31
//   B3P[(2t+h)][c] = {VN[4t+2h][c], VN[4t+2h+1][c]}      t<8, h<2, c<dim
// ---------------------------------------------------------------------------
__global__ void k_pack(const float* __restrict__ VN, float* __restrict__ BP0,
                       float* __restrict__ BP1, float* __restrict__ B3P, int dim) {
    const int idx = blockIdx.x * blockDim.x + threadIdx.x;
    const int nBP = (dim >> 2) * 32;            // float2 entries in BP0/BP1
    if (idx < nBP) {
        const int t = idx >> 5, L = idx & 31;
        const int l16 = L & 15, half = L >> 4;
        const int k = 4 * t + 2 * half;
        reinterpret_cast<float2*>(BP0)[idx] =
            make_float2(VN[(size_t)l16 * dim + k], VN[(size_t)l16 * dim + k + 1]);
        reinterpret_cast<float2*>(BP1)[idx] =
            make_float2(VN[(size_t)(l16 + 16) * dim + k],
                        VN[(size_t)(l16 + 16) * dim + k + 1]);
    }
    const int nB3 = 16 * dim;                   // float2 entries in B3P
    if (idx < nB3) {
        const int row = idx / dim;              // = 2t + h
        const int c = idx % dim;
        const int j = 2 * row;                  // VN source rows j, j+1
        reinterpret_cast<float2*>(B3P)[idx] =
            make_float2(VN[(size_t)j * dim + c], VN[(size_t)(j + 1) * dim + c]);
    }
}

// ---------------------------------------------------------------------------
// Kernel 5: fused apply. Each wave owns a 16-row tile of x.
//   S = x_tile * V          (WMMA f32 16x16x4, K = dim)
//   P = S * T               (VALU, 32x32, via LDS)
//   y = x_tile - P * V^T + bias   (WMMA f32 16x16x4, K = 32)
// ---------------------------------------------------------------------------
__global__ __launch_bounds__(256) void k_apply(
    const float* __restrict__ x, const float* __restrict__ bias,
    const float* __restrict__ BP0, const float* __restrict__ BP1,
    const float* __restrict__ B3P, const float* __restrict__ Tg,
    float* __restrict__ out, int dim) {
    __shared__ __align__(16) float Tsh[1024];
    __shared__ __align__(16) float Sbuf[8][16][32];
    __shared__ __align__(16) float Pbuf[8][16][32];

    for (int t = threadIdx.x; t < 1024; t += 256) Tsh[t] = Tg[t];
    __syncthreads();

    const int wave = threadIdx.x >> 5, lane = threadIdx.x & 31;
    const int half = lane >> 4, l16 = lane & 15;
    const int m0 = (blockIdx.x * 8 + wave) * 16;

    // ---- Stage 1: S = x_tile * V  (accumulate over K = dim) ----
    v8f s0 = {}, s1 = {};
    const float2* xrow =
        reinterpret_cast<const float2*>(x) + (((size_t)(m0 + l16) * dim) >> 1);
    const float2* bp0 = reinterpret_cast<const float2*>(BP0);
    const float2* bp1 = reinterpret_cast<const float2*>(BP1);
#pragma unroll 4
    for (int k0 = 0; k0 < dim; k0 += 4) {
        const float2 av = xrow[(k0 >> 1) + half];
        const float2 b0v = bp0[(k0 >> 2) * 32 + lane];
        const float2 b1v = bp1[(k0 >> 2) * 32 + lane];
        v2f a;  a.x = av.x;  a.y = av.y;
        v2f b0; b0.x = b0v.x; b0.y = b0v.y;
        v2f b1; b1.x = b1v.x; b1.y = b1v.y;
        s0 = __builtin_amdgcn_wmma_f32_16x16x4_f32(false, a, false, b0, (short)0,
                                                   s0, false, false);
        s1 = __builtin_amdgcn_wmma_f32_16x16x4_f32(false, a, false, b1, (short)0,
                                                   s1, false, false);
    }
    // C/D layout: acc[i] = row (i + 8*half), col l16
#pragma unroll
    for (int i = 0; i < 8; ++i) {
        Sbuf[wave][i + 8 * half][l16] = s0[i];
        Sbuf[wave][i + 8 * half][l16 + 16] = s1[i];
    }
    __syncthreads();

    // ---- Stage 2: P = S * T (tiny, VALU) ----
    {
        const float* S = &Sbuf[wave][0][0];
        float* P = &Pbuf[wave][0][0];
        for (int jj = 0; jj < 16; ++jj) {
            const int j = half * 16 + jj;
            float acc = 0.0f;
#pragma unroll 4
            for (int i = 0; i < 32; ++i) acc += S[l16 * 32 + i] * Tsh[i * 32 + j];
            P[l16 * 32 + j] = acc;
        }
    }
    __syncthreads();

    // A-fragments of P for stage 3 (K = 32 -> 8 chunks of 4)
    v2f af[8];
    {
        const float2* P2 = reinterpret_cast<const float2*>(&Pbuf[wave][0][0]);
#pragma unroll
        for (int t = 0; t < 8; ++t) {
            const float2 p = P2[l16 * 16 + t * 2 + half];
            af[t].x = p.x; af[t].y = p.y;
        }
    }

    // ---- Stage 3: y = x - P * V^T + bias ----
    const float2* b3 = reinterpret_cast<const float2*>(B3P);
    for (int c0 = 0; c0 < dim; c0 += 16) {
        v8f acc = {};
#pragma unroll
        for (int t = 0; t < 8; ++t) {
            const float2 bv = b3[(size_t)(t * 2 + half) * dim + c0 + l16];
            v2f b; b.x = bv.x; b.y = bv.y;
            acc = __builtin_amdgcn_wmma_f32_16x16x4_f32(false, af[t], false, b,
                                                        (short)0, acc, false, false);
        }
        const float bc = bias[c0 + l16];
#pragma unroll
        for (int i = 0; i < 8; ++i) {
            const size_t idx = (size_t)(m0 + i + 8 * half) * dim + c0 + l16;
            out[idx] = x[idx] - acc[i] + bc;
        }
    }
}

// ---------------------------------------------------------------------------
// Host launcher
// ---------------------------------------------------------------------------
extern "C" void kernel_launch(void* const* d_in, const int* in_sizes, int n_in,
                              void* d_out, int out_size, void* d_ws, size_t ws_size,
                              hipStream_t stream) {
    const float* x    = (const float*)d_in[0];   // [batch, dim]
    const float* vec  = (const float*)d_in[1];   // [nrefl, dim]
    const float* bias = (const float*)d_in[2];   // [dim]
    float* out = (float*)d_out;

    const int dim   = in_sizes[2];
    const int nrefl = in_sizes[1] / dim;         // 30
    const int batch = in_sizes[0] / dim;         // 16384

    // Workspace layout (floats)
    float* ws  = (float*)d_ws;
    float* VN  = ws;                              // 32*dim
    float* G   = VN + (size_t)32 * dim;           // 1024
    float* T   = G + 1024;                        // 1024
    float* BP0 = T + 1024;                        // 16*dim
    float* BP1 = BP0 + (size_t)16 * dim;          // 16*dim
    float* B3P = BP1 + (size_t)16 * dim;          // 32*dim

    k_normalize<<<32, 256, 0, stream>>>(vec, VN, dim, nrefl);
    k_gram<<<nrefl * nrefl, 256, 0, stream>>>(VN, G, dim, nrefl);
    k_buildT<<<1, 32, 0, stream>>>(G, T, nrefl);
    k_pack<<<(16 * dim + 255) / 256, 256, 0, stream>>>(VN, BP0, BP1, B3P, dim);

    // 8 waves/block, 16 rows/wave -> 128 rows per block (batch = 16384 = 128*128)
    k_apply<<<batch / 128, 256, 0, stream>>>(x, bias, BP0, BP1, B3P, T, out, dim);
}